// MultiClassNonMaxSuppression_41085657154092
// MI455X (gfx1250) — compile-verified
//
#include <hip/hip_runtime.h>

// Problem constants (match reference setup_inputs / module constants)
#define N_IMG   8
#define N_BOX   100000
#define N_CLS   80
#define CAP     4096          // per-(image,class) candidate capacity (expected ~3.1k)
#define K_TOP   100           // MAX_DET and MAX_DET_PER_CLASS
#define K_PAD   112           // 7 WMMA tiles of 16
#define IOU_THR 0.5f
#define CONF_THR 0.05f

typedef unsigned long long u64;
typedef unsigned int u32;
typedef __attribute__((ext_vector_type(2))) float v2f;
typedef __attribute__((ext_vector_type(8))) float v8f;

// ---- key packing -----------------------------------------------------------
// Softmax scores are >= 0, so raw float bits are order-preserving in the high
// 32 bits. Low 32 bits hold ~idx so that on (near-impossible) score ties the
// LOWER box index wins, matching jax.lax.top_k tie-breaking.
__device__ __forceinline__ u64 pack_key(float s, u32 idx) {
  return ((u64)__float_as_uint(s) << 32) | (u32)(~idx);
}
// Total-order encoding for possibly-negative floats (final top-k has -1.0s).
__device__ __forceinline__ u32 flip_f32(float s) {
  u32 u = __float_as_uint(s);
  return (u & 0x80000000u) ? ~u : (u | 0x80000000u);
}
__device__ __forceinline__ float unflip_f32(u32 u) {
  u = (u & 0x80000000u) ? (u & 0x7FFFFFFFu) : ~u;
  return __uint_as_float(u);
}

// ---- Phase 1: fused softmax + confidence filter (memory-bound hot loop) ----
// One thread = one box. Reads its 80 logits exactly once with b128 loads,
// computes max / exp / sum entirely in registers, appends (score, idx) keys
// above CONF_THR to per-(b,c) lists. 256 MB streamed once -> ~11 us at
// 23.3 TB/s; v_exp_f32 runs on the TRANS pipe and hides under the loads.
__device__ __forceinline__ void emit_cand(float s, int b, int c, u32 n,
                                          u64* __restrict__ cand,
                                          int* __restrict__ counts) {
  if (s > CONF_THR) {
    int bc = b * N_CLS + c;
    int pos = atomicAdd(&counts[bc], 1);
    if (pos < CAP) cand[(size_t)bc * CAP + pos] = pack_key(s, n);
  }
}

__global__ void k_softmax_filter(const float* __restrict__ conf,
                                 u64* __restrict__ cand,
                                 int* __restrict__ counts) {
  int n = blockIdx.x * blockDim.x + threadIdx.x;
  int b = blockIdx.y;
  if (n >= N_BOX) return;
  const float4* p = (const float4*)(conf + ((size_t)b * N_BOX + n) * N_CLS);
  float4 v[20];
#pragma unroll
  for (int i = 0; i < 20; ++i) v[i] = p[i];   // 20 x b128 loads, one pass
  float mx = -1e30f;
#pragma unroll
  for (int i = 0; i < 20; ++i)
    mx = fmaxf(mx, fmaxf(fmaxf(v[i].x, v[i].y), fmaxf(v[i].z, v[i].w)));
  float sum = 0.f;
#pragma unroll
  for (int i = 0; i < 20; ++i) {
    v[i].x = __expf(v[i].x - mx);
    v[i].y = __expf(v[i].y - mx);
    v[i].z = __expf(v[i].z - mx);
    v[i].w = __expf(v[i].w - mx);
    sum += v[i].x + v[i].y + v[i].z + v[i].w;
  }
  float inv = 1.0f / sum;
#pragma unroll
  for (int i = 0; i < 20; ++i) {
    emit_cand(v[i].x * inv, b, 4 * i + 0, (u32)n, cand, counts);
    emit_cand(v[i].y * inv, b, 4 * i + 1, (u32)n, cand, counts);
    emit_cand(v[i].z * inv, b, 4 * i + 2, (u32)n, cand, counts);
    emit_cand(v[i].w * inv, b, 4 * i + 3, (u32)n, cand, counts);
  }
}

// ---- Phase 2: per-(image,class) top-100 + NMS ------------------------------
// Bitonic sort of CAP u64 keys in LDS (32 KB of the 320 KB/WGP pool), then a
// 112x112 IoU matrix in 16x16 tiles. The pairwise union term
// (area_i + area_j) is a rank-2 outer-sum -> one V_WMMA_F32_16X16X4_F32 per
// tile on the matrix pipe (A = [area_i, 1 | 0 0], B = [1 ; area_j ; 0 ; 0]),
// intersection min/max on the VALU. The threshold test is division-free:
// iou > THR  <=>  inter > THR * max(area_sum - inter, 1e-8).
__global__ void k_nms(const float* __restrict__ boxes,
                      const u64* __restrict__ cand,
                      const int* __restrict__ counts,
                      float* __restrict__ kscore,
                      int* __restrict__ kidx) {
  __shared__ u64 sk[CAP];                      // 32 KB
  __shared__ float sx1[K_PAD], sy1[K_PAD], sx2[K_PAD], sy2[K_PAD];
  __shared__ float sarea[K_PAD], sscore[K_PAD];
  __shared__ int sidx[K_PAD];
  __shared__ u32 srow[K_PAD * 4];              // row -> bitmask of suppressed j>i

  int bc = blockIdx.x;
  int tid = threadIdx.x, nthr = blockDim.x;
  int cnt = counts[bc];
  if (cnt > CAP) cnt = CAP;

  for (int i = tid; i < CAP; i += nthr)
    sk[i] = (i < cnt) ? cand[(size_t)bc * CAP + i] : 0ull;
  for (int i = tid; i < K_PAD * 4; i += nthr) srow[i] = 0u;
  __syncthreads();

  // Bitonic sort, descending (keys unique -> deterministic result).
  for (int k = 2; k <= CAP; k <<= 1) {
    for (int j = k >> 1; j > 0; j >>= 1) {
      __syncthreads();
      for (int i = tid; i < CAP; i += nthr) {
        int ixj = i ^ j;
        if (ixj > i) {
          u64 a = sk[i], b2 = sk[ixj];
          bool desc = ((i & k) == 0);
          if (desc ? (a < b2) : (a > b2)) { sk[i] = b2; sk[ixj] = a; }
        }
      }
    }
  }
  __syncthreads();

  int b = bc / N_CLS;
  for (int i = tid; i < K_PAD; i += nthr) {
    bool real = (i < K_TOP) && (i < cnt);
    if (real) {
      u64 key = sk[i];
      int idx = (int)(~(u32)key);
      const float* bp = boxes + ((size_t)b * N_BOX + idx) * 4;
      float x1 = bp[0], y1 = bp[1], x2 = bp[2], y2 = bp[3];
      sx1[i] = x1; sy1[i] = y1; sx2[i] = x2; sy2[i] = y2;
      sarea[i] = (x2 - x1) * (y2 - y1);
      sscore[i] = __uint_as_float((u32)(key >> 32));
      sidx[i] = idx;
    } else {                                   // degenerate pad: IoU == 0
      sx1[i] = sy1[i] = sx2[i] = sy2[i] = 0.f;
      sarea[i] = 0.f; sscore[i] = 0.f; sidx[i] = 0;
    }
  }
  __syncthreads();

  // 7x7 tiles; tile index is wave-uniform so EXEC is all-ones at the WMMA.
  int lane = tid & 31, wv = tid >> 5, nwv = nthr >> 5;
  int m = lane & 15;
  bool lo = lane < 16;
  for (int t = wv; t < 49; t += nwv) {
    int ti = t / 7, tj = t % 7;
    v2f a, bb;
    v8f d = {};
    // A 16x4 f32: lanes 0-15 carry K0 (VGPR0), K1 (VGPR1); lanes 16-31 (K2,K3)=0
    a[0] = lo ? sarea[ti * 16 + m] : 0.f;
    a[1] = lo ? 1.f : 0.f;
    bb[0] = lo ? 1.f : 0.f;
    bb[1] = lo ? sarea[tj * 16 + m] : 0.f;
    d = __builtin_amdgcn_wmma_f32_16x16x4_f32(false, a, false, bb,
                                              (short)0, d, false, false);
    // D layout: VGPR r -> (M = ti*16 + r + (lane>=16 ? 8 : 0), N = tj*16 + m)
    int i0 = ti * 16 + (lo ? 0 : 8);
    int jj = tj * 16 + m;
    float jx1 = sx1[jj], jy1 = sy1[jj], jx2 = sx2[jj], jy2 = sy2[jj];
#pragma unroll
    for (int r = 0; r < 8; ++r) {
      int i = i0 + r;
      float iw = fminf(sx2[i], jx2) - fmaxf(sx1[i], jx1);
      float ih = fminf(sy2[i], jy2) - fmaxf(sy1[i], jy1);
      float inter = fmaxf(iw, 0.f) * fmaxf(ih, 0.f);
      float uni = fmaxf(d[r] - inter, 1e-8f);   // area_i + area_j from WMMA
      // inter/uni > IOU_THR  <=>  inter > IOU_THR*uni   (uni > 0)
      if (i < K_TOP && jj < K_TOP && jj > i && inter > IOU_THR * uni)
        atomicOr(&srow[i * 4 + (jj >> 5)], 1u << (jj & 31));
    }
  }
  __syncthreads();

  // Greedy scan (100 cheap serial steps on one lane).
  if (tid == 0) {
    u32 sup[4] = {0u, 0u, 0u, 0u};
    for (int i = 0; i < K_TOP; ++i) {
      bool suppressed = (sup[i >> 5] >> (i & 31)) & 1u;
      bool keep = !suppressed && (sscore[i] > CONF_THR);
      if (keep) {
        sup[0] |= srow[i * 4 + 0]; sup[1] |= srow[i * 4 + 1];
        sup[2] |= srow[i * 4 + 2]; sup[3] |= srow[i * 4 + 3];
      }
      kscore[(size_t)bc * K_TOP + i] = keep ? sscore[i] : -1.0f;
      kidx[(size_t)bc * K_TOP + i] = sidx[i];
    }
  }
}

// ---- Phase 3: per-image top-100 over 80*100 kept scores --------------------
// 8000 flip-encoded u64 keys in LDS (62.5 KB). 100x block argmax with wave32
// shfl reductions; thread 0 writes boxes/score/class and counts valid dets.
#define FLAT (N_CLS * K_TOP)
__global__ void k_final(const float* __restrict__ boxes,
                        const float* __restrict__ kscore,
                        const int* __restrict__ kidx,
                        float* __restrict__ out) {
  __shared__ u64 fk[FLAT];
  __shared__ u64 wred[8];
  __shared__ u64 winner;
  int b = blockIdx.x, tid = threadIdx.x, nthr = blockDim.x;
  int lane = tid & 31, wv = tid >> 5, nwv = nthr >> 5;

  for (int i = tid; i < FLAT; i += nthr) {
    float s = kscore[(size_t)b * FLAT + i];
    fk[i] = ((u64)flip_f32(s) << 32) | (u32)(~(u32)i); // low idx wins ties
  }
  __syncthreads();

  float* ob = out;                           // (8,100,4)
  float* os = out + N_IMG * K_TOP * 4;       // (8,100)
  float* oc = os + N_IMG * K_TOP;            // (8,100)
  float* ond = oc + N_IMG * K_TOP;           // (8,)
  int nd = 0;

  for (int r = 0; r < K_TOP; ++r) {
    u64 lbest = 0; int lpos = -1;
    for (int i = tid; i < FLAT; i += nthr)
      if (fk[i] > lbest) { lbest = fk[i]; lpos = i; }
    u64 rb = lbest;
    for (int off = 16; off > 0; off >>= 1) {  // wave32 reduction
      u64 o = __shfl_down(rb, off);
      if (o > rb) rb = o;
    }
    if (lane == 0) wred[wv] = rb;
    __syncthreads();
    if (tid == 0) {
      u64 w = wred[0];
      for (int q = 1; q < nwv; ++q) if (wred[q] > w) w = wred[q];
      winner = w;
    }
    __syncthreads();
    u64 w = winner;
    if (lpos >= 0 && lbest == w) fk[lpos] = 0; // unique keys -> one owner
    if (tid == 0) {
      u32 fi = ~(u32)w;                        // flat index in [0, 8000)
      float s = unflip_f32((u32)(w >> 32));
      int o = b * K_TOP + r;
      if (s > -0.5f) {
        int cls = fi / K_TOP;
        int bi = kidx[(size_t)b * FLAT + fi];
        const float* bp = boxes + ((size_t)b * N_BOX + bi) * 4;
        ob[o * 4 + 0] = bp[0]; ob[o * 4 + 1] = bp[1];
        ob[o * 4 + 2] = bp[2]; ob[o * 4 + 3] = bp[3];
        os[o] = s; oc[o] = (float)cls;
        nd++;
      } else {
        ob[o * 4 + 0] = -1.f; ob[o * 4 + 1] = -1.f;
        ob[o * 4 + 2] = -1.f; ob[o * 4 + 3] = -1.f;
        os[o] = -1.f; oc[o] = -1.f;
      }
    }
    __syncthreads();                           // clears visible before rescan
  }
  if (tid == 0) ond[b] = (float)nd;
}

// ---- launch ----------------------------------------------------------------
extern "C" void kernel_launch(void* const* d_in, const int* in_sizes, int n_in,
                              void* d_out, int out_size, void* d_ws, size_t ws_size,
                              hipStream_t stream) {
  const float* boxes = (const float*)d_in[0];  // (8,100000,4) xyxy
  const float* conf  = (const float*)d_in[1];  // (8,100000,80) logits
  float* out = (float*)d_out;                  // 3200 + 800 + 800 + 8 floats

  char* ws = (char*)d_ws;
  int* counts = (int*)ws;                                        // 2560 B
  u64* cand = (u64*)(ws + 4096);                                 // 20 MB
  size_t cand_bytes = (size_t)N_IMG * N_CLS * CAP * sizeof(u64);
  float* kscore = (float*)(ws + 4096 + cand_bytes);              // 256 KB
  int* kidx = (int*)(ws + 4096 + cand_bytes +
                     (size_t)N_IMG * N_CLS * K_TOP * sizeof(float));

  hipMemsetAsync(counts, 0, N_IMG * N_CLS * sizeof(int), stream);

  dim3 g1((N_BOX + 255) / 256, N_IMG);
  k_softmax_filter<<<g1, 256, 0, stream>>>(conf, cand, counts);
  k_nms<<<N_IMG * N_CLS, 256, 0, stream>>>(boxes, cand, counts, kscore, kidx);
  k_final<<<N_IMG, 256, 0, stream>>>(boxes, kscore, kidx, out);
}